// DynamicGraph_55147380081147
// MI455X (gfx1250) — compile-verified
//
#include <hip/hip_runtime.h>

typedef __attribute__((ext_vector_type(2))) float v2f;
typedef __attribute__((ext_vector_type(8))) float v8f;

#define DHID     256
#define LTOK     128
#define LLAB     32
#define NPG      64
#define BGRAPH   64
#define NNODES   (BGRAPH * NPG)
#define ODIM     1024

// ---------------------------------------------------------------------------
// Kernel 1: EmbeddingBag-style ragged gather.
// One block per node, 256 threads = one thread per channel d.
// Each word_emb/label_emb row read is a coalesced 1KB burst; both tables are
// L2-resident (51MB + 102MB < 192MB L2), so this runs at L2 gather bandwidth.
// ---------------------------------------------------------------------------
__global__ void ragged_bag_kernel(const int* __restrict__ text_ids,
                                  const int* __restrict__ label_ids,
                                  const int* __restrict__ text_len,
                                  const int* __restrict__ label_len,
                                  const float* __restrict__ word_emb,
                                  const float* __restrict__ label_emb,
                                  float* __restrict__ text_out,
                                  float* __restrict__ label_out) {
    __shared__ int s_tid[LTOK];
    __shared__ int s_lid[LLAB];
    const int n = blockIdx.x;
    const int d = threadIdx.x;

    if (threadIdx.x < LTOK) s_tid[threadIdx.x] = text_ids[n * LTOK + threadIdx.x];
    if (threadIdx.x < LLAB) s_lid[threadIdx.x] = label_ids[n * LLAB + threadIdx.x];
    __syncthreads();

    const int tl = text_len[n];
    float acc = 0.0f;
    for (int l = 0; l < tl; ++l)
        acc += word_emb[(size_t)s_tid[l] * DHID + d];
    text_out[(size_t)n * DHID + d] = acc / (float)tl;

    const int ll = label_len[n];
    float accl = 0.0f;
    for (int l = 0; l < ll; ++l)
        accl += label_emb[(size_t)s_lid[l] * DHID + d];
    label_out[(size_t)n * DHID + d] = accl;
}

// ---------------------------------------------------------------------------
// Kernel 2: generic fp32 GEMM C = act(A[MxK] * B[KxN] + bias) using
// V_WMMA_F32_16X16X4_F32. 8 waves / 256-thread block; each wave owns one
// 16x16 output tile and marches K in steps of 4.
//
// wave32 VGPR layouts (ISA 7.12.2):
//   A 16x4:  lane l -> row m=l&15, a[j] = A[m][2*(l>>4)+j]
//   B 4x16:  lane l -> col n=l&15, b[j] = B[2*(l>>4)+j][n]
//   C 16x16: lane l -> col n=l&15, c[v] = C[(l>>4)*8 + v][n]
// Tile guard is wave-uniform, so EXEC is all-ones at every WMMA.
// ---------------------------------------------------------------------------
__global__ void gemm_wmma_f32(const float* __restrict__ A,
                              const float* __restrict__ B,
                              const float* __restrict__ bias,
                              float* __restrict__ C,
                              int M, int N, int K, int relu) {
    const int wave       = (blockIdx.x * blockDim.x + threadIdx.x) >> 5;
    const int tilesN     = N >> 4;
    const int totalTiles = (M >> 4) * tilesN;
    if (wave >= totalTiles) return;   // uniform per wave

    const int tm   = wave / tilesN;
    const int tn   = wave % tilesN;
    const int lane = threadIdx.x & 31;
    const int kh   = lane >> 4;       // which K-pair this half-wave owns
    const int l15  = lane & 15;

    const float* Arow = A + (size_t)(tm * 16 + l15) * K;   // row m of A
    const float* Bcol = B + (size_t)(tn * 16 + l15);       // col n of B

    v8f acc = {};
    for (int k = 0; k < K; k += 4) {
        const int kk = k + kh * 2;
        v2f a = *(const v2f*)(Arow + kk);                  // A[m][kk], A[m][kk+1]
        v2f b;
        b.x = Bcol[(size_t)kk * N];                        // B[kk][n]
        b.y = Bcol[(size_t)(kk + 1) * N];                  // B[kk+1][n]
        acc = __builtin_amdgcn_wmma_f32_16x16x4_f32(
            /*neg_a=*/false, a, /*neg_b=*/false, b,
            /*c_mod=*/(short)0, acc, /*reuse_a=*/false, /*reuse_b=*/false);
    }

    const int col  = tn * 16 + l15;
    const int row0 = tm * 16 + kh * 8;
    const float bv = bias ? bias[col] : 0.0f;
#pragma unroll
    for (int v = 0; v < 8; ++v) {
        float val = acc[v] + bv;
        if (relu) val = fmaxf(val, 0.0f);
        C[(size_t)(row0 + v) * N + col] = val;
    }
}

// ---------------------------------------------------------------------------
// Kernel 3: per-graph mean readout of h_t/h_l + adaptive fusion with w_adap.
// One block per graph, one thread per channel.
// ---------------------------------------------------------------------------
__global__ void readout_kernel(const float* __restrict__ h_t,
                               const float* __restrict__ h_l,
                               const float* __restrict__ w_adap,
                               float* __restrict__ fused) {
    const int b = blockIdx.x;
    const int d = threadIdx.x;
    float st = 0.0f, sl = 0.0f;
    for (int i = 0; i < NPG; ++i) {
        const size_t node = (size_t)b * NPG + i;
        st += h_t[node * DHID + d];
        sl += h_l[node * DHID + d];
    }
    const float inv = 1.0f / (float)NPG;
    fused[(size_t)b * DHID + d] = (st * inv) * w_adap[0] + (sl * inv) * w_adap[1];
}

// ---------------------------------------------------------------------------
extern "C" void kernel_launch(void* const* d_in, const int* in_sizes, int n_in,
                              void* d_out, int out_size, void* d_ws, size_t ws_size,
                              hipStream_t stream) {
    const int*   text_ids  = (const int*)  d_in[0];
    const int*   label_ids = (const int*)  d_in[1];
    const int*   text_len  = (const int*)  d_in[2];
    const int*   label_len = (const int*)  d_in[3];
    const float* word_emb  = (const float*)d_in[4];
    const float* label_emb = (const float*)d_in[5];
    const float* Wg_text   = (const float*)d_in[6];
    const float* Wg_label  = (const float*)d_in[7];
    const float* w_adap    = (const float*)d_in[8];
    const float* W1        = (const float*)d_in[9];
    const float* b1        = (const float*)d_in[10];
    const float* Wout      = (const float*)d_in[11];
    const float* bout      = (const float*)d_in[12];
    float* out = (float*)d_out;

    // Workspace layout (floats): ~17 MB total
    float* ws        = (float*)d_ws;
    float* text_out  = ws;                                   // [NNODES, DHID]
    float* label_out = text_out  + (size_t)NNODES * DHID;    // [NNODES, DHID]
    float* h_t       = label_out + (size_t)NNODES * DHID;    // [NNODES, DHID]
    float* h_l       = h_t       + (size_t)NNODES * DHID;    // [NNODES, DHID]
    float* fused     = h_l       + (size_t)NNODES * DHID;    // [BGRAPH, DHID]
    float* hmid      = fused     + (size_t)BGRAPH * DHID;    // [BGRAPH, DHID]

    // 1) ragged embedding bags
    ragged_bag_kernel<<<NNODES, 256, 0, stream>>>(
        text_ids, label_ids, text_len, label_len,
        word_emb, label_emb, text_out, label_out);

    // 2) per-type projection + relu (WMMA f32)
    const int tiles1 = (NNODES / 16) * (DHID / 16);
    gemm_wmma_f32<<<(tiles1 + 7) / 8, 256, 0, stream>>>(
        text_out, Wg_text, nullptr, h_t, NNODES, DHID, DHID, /*relu=*/1);
    gemm_wmma_f32<<<(tiles1 + 7) / 8, 256, 0, stream>>>(
        label_out, Wg_label, nullptr, h_l, NNODES, DHID, DHID, /*relu=*/1);

    // 3) per-graph mean readout + adaptive fusion
    readout_kernel<<<BGRAPH, 256, 0, stream>>>(h_t, h_l, w_adap, fused);

    // 4) h = relu(fused @ W1 + b1)
    const int tiles2 = (BGRAPH / 16) * (DHID / 16);
    gemm_wmma_f32<<<(tiles2 + 7) / 8, 256, 0, stream>>>(
        fused, W1, b1, hmid, BGRAPH, DHID, DHID, /*relu=*/1);

    // 5) out = h @ Wout + bout
    const int tiles3 = (BGRAPH / 16) * (ODIM / 16);
    gemm_wmma_f32<<<(tiles3 + 7) / 8, 256, 0, stream>>>(
        hmid, Wout, bout, out, BGRAPH, ODIM, DHID, /*relu=*/0);
}